// ResGatedGraphConv_86285892976710
// MI455X (gfx1250) — compile-verified
//
#include <hip/hip_runtime.h>

#define NN 50000
#define NE 800000
#define DD 128

typedef float v2f __attribute__((ext_vector_type(2)));
typedef float v8f __attribute__((ext_vector_type(8)));

// ---------------------------------------------------------------------------
// Kernel 1: five fused GEMMs  Hk = X @ Wk.T + bk  using V_WMMA_F32_16X16X4_F32
// Block = 256 threads = 8 waves. Block handles a 16-row tile of X (staged in
// LDS, padded stride 132 to kill bank conflicts). Wave w computes output
// columns [16w, 16w+16) for ALL five weights, reusing the A fragment 5x.
// ---------------------------------------------------------------------------
__global__ __launch_bounds__(256) void gemm5_wmma(
    const float* __restrict__ X,
    const float* __restrict__ W1, const float* __restrict__ b1,
    const float* __restrict__ W2, const float* __restrict__ b2,
    const float* __restrict__ W3, const float* __restrict__ b3,
    const float* __restrict__ W4, const float* __restrict__ b4,
    const float* __restrict__ Wr, const float* __restrict__ br,
    float* __restrict__ H1, float* __restrict__ H2,
    float* __restrict__ H3, float* __restrict__ H4,
    float* __restrict__ Hr)
{
    __shared__ float xs[16 * 132];
    const int row0 = blockIdx.x * 16;
    const int t = threadIdx.x;

    // cooperative load of 16x128 fp32 tile (8 KB) -> LDS, padded rows
    #pragma unroll
    for (int q = t; q < 512; q += 256) {
        const int r  = q >> 5;
        const int c4 = (q & 31) << 2;
        const float4 v = *reinterpret_cast<const float4*>(X + (size_t)(row0 + r) * DD + c4);
        *reinterpret_cast<float4*>(&xs[r * 132 + c4]) = v;
    }
    __syncthreads();

    const int wave = t >> 5;
    const int lane = t & 31;
    const int half = lane >> 4;   // 0: lanes 0-15, 1: lanes 16-31
    const int nn   = lane & 15;   // row (A) / column (B,D) index within tile
    const int n0   = wave * 16;   // this wave's output-column tile
    const int wrow = n0 + nn;     // row of W == output column

    const float* Ws[5] = {W1, W2, W3, W4, Wr};
    const float* Bs[5] = {b1, b2, b3, b4, br};
    float*       Hs[5] = {H1, H2, H3, H4, Hr};

    v8f acc[5] = {};

    for (int k = 0; k < DD; k += 4) {
        const int kk = k + 2 * half; // lane's K pair within this 4-wide step
        // A fragment: X[m = nn, kk..kk+1] from LDS (b64, conflict-free)
        const float2 af = *reinterpret_cast<const float2*>(&xs[nn * 132 + kk]);
        v2f a; a.x = af.x; a.y = af.y;
        #pragma unroll
        for (int w = 0; w < 5; ++w) {
            // B fragment: (Wk.T)[kk..kk+1, n = nn] == Wk[wrow, kk..kk+1]
            const float2 bf = *reinterpret_cast<const float2*>(Ws[w] + (size_t)wrow * DD + kk);
            v2f b; b.x = bf.x; b.y = bf.y;
            acc[w] = __builtin_amdgcn_wmma_f32_16x16x4_f32(
                false, a, false, b, (short)0, acc[w], false, false);
        }
    }

    // D layout: VGPR j holds row m = j + 8*half, column n = nn (per half-wave)
    #pragma unroll
    for (int w = 0; w < 5; ++w) {
        const float bias = Bs[w][wrow];
        #pragma unroll
        for (int j = 0; j < 8; ++j) {
            const int m = j + 8 * half;
            Hs[w][(size_t)(row0 + m) * DD + wrow] = acc[w][j] + bias;
        }
    }
}

// ---------------------------------------------------------------------------
// Kernel 2: edge phase. One wave32 per edge; 4 floats per lane (float4 =
// global_load_b128, fully coalesced: 3 x 512B gathers per edge). Gate with
// sigmoid, scatter-add gated message into H1 (which already holds h1+b1)
// via hardware global_atomic_add_f32; lane 0 bumps the degree counter.
// ---------------------------------------------------------------------------
__global__ __launch_bounds__(256) void edge_gate_scatter(
    const int* __restrict__ ei,
    const float* __restrict__ H2, const float* __restrict__ H3,
    const float* __restrict__ H4,
    float* __restrict__ H1acc, float* __restrict__ cnt)
{
    const int wave = threadIdx.x >> 5;
    const int lane = threadIdx.x & 31;
    const int e = blockIdx.x * 8 + wave;
    if (e >= NE) return;

    const int src = ei[e];        // edge_index[0, e]
    const int dst = ei[NE + e];   // edge_index[1, e]
    const int d0 = lane << 2;

    const float4 x3 = *reinterpret_cast<const float4*>(H3 + (size_t)src * DD + d0);
    const float4 x4 = *reinterpret_cast<const float4*>(H4 + (size_t)dst * DD + d0);
    const float4 x2 = *reinterpret_cast<const float4*>(H2 + (size_t)src * DD + d0);

    float4 m;
    m.x = x2.x / (1.0f + __expf(-(x3.x + x4.x)));
    m.y = x2.y / (1.0f + __expf(-(x3.y + x4.y)));
    m.z = x2.z / (1.0f + __expf(-(x3.z + x4.z)));
    m.w = x2.w / (1.0f + __expf(-(x3.w + x4.w)));

    float* p = H1acc + (size_t)dst * DD + d0;
    unsafeAtomicAdd(p + 0, m.x);
    unsafeAtomicAdd(p + 1, m.y);
    unsafeAtomicAdd(p + 2, m.z);
    unsafeAtomicAdd(p + 3, m.w);

    if (lane == 0) unsafeAtomicAdd(&cnt[dst], 1.0f);
}

// ---------------------------------------------------------------------------
// Kernel 3: out = (h1 + sum_msgs)/max(cnt,1) + hres   (hres already in d_out)
// float4 per thread; 6.4M floats -> 1.6M float4 -> 6250 blocks.
// ---------------------------------------------------------------------------
__global__ __launch_bounds__(256) void finalize_mean_res(
    const float* __restrict__ H1acc, const float* __restrict__ cnt,
    float* __restrict__ out)
{
    const size_t i = (size_t)blockIdx.x * 256 + threadIdx.x; // float4 index
    const int node = (int)(i >> 5);                          // 32 float4 / node
    const float s = 1.0f / fmaxf(cnt[node], 1.0f);
    const float4 h = reinterpret_cast<const float4*>(H1acc)[i];
    float4 o = reinterpret_cast<float4*>(out)[i];
    o.x += h.x * s;
    o.y += h.y * s;
    o.z += h.z * s;
    o.w += h.w * s;
    reinterpret_cast<float4*>(out)[i] = o;
}

// ---------------------------------------------------------------------------
extern "C" void kernel_launch(void* const* d_in, const int* in_sizes, int n_in,
                              void* d_out, int out_size, void* d_ws, size_t ws_size,
                              hipStream_t stream)
{
    const float* X   = (const float*)d_in[0];
    const int*   ei  = (const int*)  d_in[1];
    const float* W1  = (const float*)d_in[2];
    const float* b1  = (const float*)d_in[3];
    const float* W2  = (const float*)d_in[4];
    const float* b2  = (const float*)d_in[5];
    const float* W3  = (const float*)d_in[6];
    const float* b3  = (const float*)d_in[7];
    const float* W4  = (const float*)d_in[8];
    const float* b4  = (const float*)d_in[9];
    const float* Wr  = (const float*)d_in[10];
    const float* br  = (const float*)d_in[11];
    float* out = (float*)d_out;

    const size_t nd = (size_t)NN * DD;
    float* ws  = (float*)d_ws;
    float* H1  = ws;            // h1 + b1, then += gated messages (atomics)
    float* H2  = ws + nd;
    float* H3  = ws + 2 * nd;
    float* H4  = ws + 3 * nd;
    float* cnt = ws + 4 * nd;   // NN floats

    hipMemsetAsync(cnt, 0, NN * sizeof(float), stream);

    gemm5_wmma<<<NN / 16, 256, 0, stream>>>(
        X, W1, b1, W2, b2, W3, b3, W4, b4, Wr, br,
        H1, H2, H3, H4, /*Hres -> */ out);

    edge_gate_scatter<<<NE / 8, 256, 0, stream>>>(ei, H2, H3, H4, H1, cnt);

    finalize_mean_res<<<(int)(nd / 4 / 256), 256, 0, stream>>>(H1, cnt, out);
}